// WENONetwork_19963007991813
// MI455X (gfx1250) — compile-verified
//
#include <hip/hip_runtime.h>
#include <math.h>

// ---------------------------------------------------------------------------
// WENO5 Black-Scholes digital option, persistent single-WGP kernel.
// State (1025 f32) lives in LDS for all 2098*3 RK stages; one s_barrier per
// stage; global<->LDS movement uses the CDNA5 async-LDS DMA path
// (global_load_async_to_lds_b32 / global_store_async_from_lds_b32, ASYNCcnt).
// Latency-bound problem: 6294 dependent stages, so the design minimizes
// per-stage critical path (trivial BC points, 16 wave32 waves to hide the
// FP32 divide latency, single barrier per stage).
// ---------------------------------------------------------------------------

#define N_GRID 1025
#define EXT    (N_GRID + 6)   // 3 replicated ghost cells per side
#define BS     512            // 16 wave32 waves on one WGP (4 per SIMD32)

#if defined(__has_builtin)
#  if __has_builtin(__builtin_amdgcn_global_load_async_to_lds_b32)
#    define HAS_ASYNC_LD 1
#  endif
#  if __has_builtin(__builtin_amdgcn_global_store_async_from_lds_b32)
#    define HAS_ASYNC_ST 1
#  endif
#  if __has_builtin(__builtin_amdgcn_s_wait_asynccnt)
#    define HAS_WAIT_ASYNC 1
#  endif
#endif

// Address-space-qualified pointer types the async builtins expect.
typedef __attribute__((address_space(1))) int GInt;   // global
typedef __attribute__((address_space(3))) int LInt;   // LDS

__device__ __forceinline__ void wait_asynccnt0() {
#ifdef HAS_WAIT_ASYNC
  __builtin_amdgcn_s_wait_asynccnt(0);
#else
  asm volatile("s_wait_asynccnt 0" ::: "memory");
#endif
}

// Classical WENO5 reconstruction (matches reference formulas; eps = 1e-13).
__device__ __forceinline__ float weno5(float fm2, float fm1, float f0,
                                       float fp1, float fp2) {
  const float EPSW = 1e-13f;
  float d0  = fm2 - 2.0f * fm1 + f0;
  float d0b = fm2 - 4.0f * fm1 + 3.0f * f0;
  float b0  = (13.0f / 12.0f) * d0 * d0 + 0.25f * d0b * d0b;
  float d1  = fm1 - 2.0f * f0 + fp1;
  float d1b = fm1 - fp1;
  float b1  = (13.0f / 12.0f) * d1 * d1 + 0.25f * d1b * d1b;
  float d2  = f0 - 2.0f * fp1 + fp2;
  float d2b = 3.0f * f0 - 4.0f * fp1 + fp2;
  float b2  = (13.0f / 12.0f) * d2 * d2 + 0.25f * d2b * d2b;
  float t0 = EPSW + b0, t1 = EPSW + b1, t2 = EPSW + b2;
  float a0 = 0.1f / (t0 * t0);
  float a1 = 0.6f / (t1 * t1);
  float a2 = 0.3f / (t2 * t2);
  const float SIXTH = 1.0f / 6.0f;
  float q0 = (2.0f * fm2 - 7.0f * fm1 + 11.0f * f0) * SIXTH;
  float q1 = (-fm1 + 5.0f * f0 + 2.0f * fp1) * SIXTH;
  float q2 = (2.0f * f0 + 5.0f * fp1 - fp2) * SIXTH;
  return (a0 * q0 + a1 * q1 + a2 * q2) / (a0 + a1 + a2);
}

// One SSP-RK3 stage: v = ca*u^n + cb*(u_in + tau*rhs(u_in)), then BC.
// S==0 additionally snapshots u^n into sU0. BC points skip all flux work
// (their values are overwritten by the Dirichlet BC anyway).
template <int S>
__device__ __forceinline__ void rk_stage(const float* __restrict__ cur,
                                         float* __restrict__ nxt,
                                         float* __restrict__ sU0,
                                         int tid, float tau, float bcval) {
  const float ca = (S == 0) ? 0.0f : (S == 1 ? 0.75f : (1.0f / 3.0f));
  const float cb = (S == 0) ? 1.0f : (S == 1 ? 0.25f : (2.0f / 3.0f));
  const float c1      = 0.045f;                 // 0.5*sigma^2
  const float c2      = 0.055f;                 // rate - c1  (>0 -> fm flux == 0)
  const float rate    = 0.1f;
  const float invH    = (float)(1024.0 / 7.5);
  const float inv12h2 = (float)(1.0 / (12.0 * (7.5 / 1024.0) * (7.5 / 1024.0)));

  for (int j = tid; j < N_GRID; j += BS) {
    if (j == 0) {                       // Dirichlet left: v = 0, no flux work
      nxt[0] = 0.0f; nxt[1] = 0.0f; nxt[2] = 0.0f; nxt[3] = 0.0f;
      continue;
    }
    if (j == N_GRID - 1) {              // discounted cash right, no flux work
      nxt[j + 3] = bcval;
      nxt[EXT - 3] = bcval; nxt[EXT - 2] = bcval; nxt[EXT - 1] = bcval;
      continue;
    }

    // extended-index window e[j .. j+5]
    float em2 = cur[j];
    float em1 = cur[j + 1];
    float e0  = cur[j + 2];
    float e1  = cur[j + 3];   // u_j
    float e2  = cur[j + 4];
    float e3  = cur[j + 5];

    if (S == 0) sU0[j] = e1;  // snapshot u^n for later RK combinations

    float g0 = c2 * em2, g1 = c2 * em1, g2 = c2 * e0;
    float g3 = c2 * e1,  g4 = c2 * e2,  g5 = c2 * e3;

    // fm == 0 exactly (c2 > 0): only the left-biased flux survives.
    float fhL = weno5(g0, g1, g2, g3, g4);
    float fhR = weno5(g1, g2, g3, g4, g5);

    float uxx = (-em1 + 16.0f * e0 - 30.0f * e1 + 16.0f * e2 - e3) * inv12h2;
    float rhs = c1 * uxx + (fhR - fhL) * invH - rate * e1;

    float stage = e1 + tau * rhs;
    float v = (S == 0) ? stage : (ca * sU0[j] + cb * stage);
    nxt[j + 3] = v;
  }
}

__global__ __launch_bounds__(BS) void weno_bs_kernel(const float* __restrict__ u0,
                                                     float* __restrict__ out,
                                                     int nsteps, float tau) {
  __shared__ float sA[EXT];       // ping buffer (extended with ghosts)
  __shared__ float sB[EXT];       // pong buffer
  __shared__ float sU0[N_GRID];   // RK step-start state u^n

  const int tid = threadIdx.x;
  const float rate = 0.1f;

  // ---- load u0 -> LDS (async DMA path) ----
#ifdef HAS_ASYNC_LD
  for (int j = tid; j < N_GRID; j += BS) {
    __builtin_amdgcn_global_load_async_to_lds_b32(
        (GInt*)(u0 + j), (LInt*)(&sA[3 + j]), 0, 0);
  }
  wait_asynccnt0();
#else
  for (int j = tid; j < N_GRID; j += BS) sA[3 + j] = u0[j];
#endif
  __syncthreads();
  if (tid == 0) {
    float l = sA[3], r = sA[3 + N_GRID - 1];
    sA[0] = sA[1] = sA[2] = l;
    sA[EXT - 3] = sA[EXT - 2] = sA[EXT - 1] = r;
  }
  __syncthreads();

  float* cur = sA;
  float* nxt = sB;
  float t = 0.0f;

  for (int step = 0; step < nsteps; ++step) {
    float tn    = t + tau;
    float bcval = expf(-rate * tn) * 0.02f;   // exp(-r*t)/E, E = 50

    rk_stage<0>(cur, nxt, sU0, tid, tau, bcval);
    __syncthreads();
    { float* tmp = cur; cur = nxt; nxt = tmp; }

    rk_stage<1>(cur, nxt, sU0, tid, tau, bcval);
    __syncthreads();
    { float* tmp = cur; cur = nxt; nxt = tmp; }

    rk_stage<2>(cur, nxt, sU0, tid, tau, bcval);
    __syncthreads();
    { float* tmp = cur; cur = nxt; nxt = tmp; }

    t = tn;
  }

  // ---- scale by E and store (async LDS->global DMA path) ----
#ifdef HAS_ASYNC_ST
  for (int j = tid; j < N_GRID; j += BS) nxt[j] = cur[j + 3] * 50.0f;
  __syncthreads();                      // make LDS staging visible to the DMA
  for (int j = tid; j < N_GRID; j += BS) {
    __builtin_amdgcn_global_store_async_from_lds_b32(
        (GInt*)(out + j), (LInt*)(&nxt[j]), 0, 0);
  }
  wait_asynccnt0();
#else
  for (int j = tid; j < N_GRID; j += BS) out[j] = cur[j + 3] * 50.0f;
#endif
}

extern "C" void kernel_launch(void* const* d_in, const int* in_sizes, int n_in,
                              void* d_out, int out_size, void* d_ws, size_t ws_size,
                              hipStream_t stream) {
  (void)in_sizes; (void)n_in; (void)out_size; (void)d_ws; (void)ws_size;
  const float* u0  = (const float*)d_in[0];
  float*       out = (float*)d_out;

  // Mirror the reference's step-count computation in double precision.
  const double H      = 7.5 / 1024.0;
  const double tau0   = 0.8 * H * H / (0.3 * 0.3);
  const int    nsteps = (int)ceil(1.0 / tau0);        // = 2098
  const float  tau    = (float)(1.0 / (double)nsteps);

  weno_bs_kernel<<<dim3(1), dim3(BS), 0, stream>>>(u0, out, nsteps, tau);
}